// Model_VAE_42133629174232
// MI455X (gfx1250) — compile-verified
//
#include <hip/hip_runtime.h>

// CDNA5 / gfx1250: wave32, V_WMMA_F32_16X16X4_F32 (D = A(16x4) x B(4x16) + C).
// One wave = one batch item; all matrices padded to 16x16 f32 tiles.
// A prep kernel packs all constant operands into d_ws in per-lane fragment
// layout so the hot kernel has zero guarded loads.

typedef __attribute__((ext_vector_type(2))) float v2f;
typedef __attribute__((ext_vector_type(4))) float v4f;
typedef __attribute__((ext_vector_type(8))) float v8f;

#define BN_EPS 1e-5f

// workspace layout (floats)
enum {
    WB_INIT = 0,        // 256  : padded init_weight B-frags
    WB_W0   = 256,      // 1024 : w0[l]^T B-frags, l=0..3
    WB_W1   = 1280,     // 1024 : w1[l]^T B-frags
    WB_FC1  = 2304,     // 256
    WB_FC2  = 2560,     // 256
    WB_DEC  = 2816,     // 1024 : dec_w[k]^T B-frags
    BN_SIN  = 3840,     // 64   : BN-in scale  (rows 13..15 = 0)
    BN_TIN  = 3904,     // 64   : BN-in offset
    BN_SOUT = 3968,     // 64
    BN_TOUT = 4032,     // 64
    WS_TOTAL = 4096     // 16 KB
};

static __device__ __forceinline__ v8f wmma4(v2f a, v2f b, v8f c) {
    // 8 args: (neg_a, A, neg_b, B, c_mod, C, reuse_a, reuse_b)
    return __builtin_amdgcn_wmma_f32_16x16x4_f32(false, a, false, b, (short)0, c,
                                                 false, false);
}

static __device__ __forceinline__ v8f zero8() {
    v8f z = {0.f, 0.f, 0.f, 0.f, 0.f, 0.f, 0.f, 0.f};
    return z;
}

// pick K-step kk fragment out of a lane's packed 8 floats (kk is unrolled-const)
static __device__ __forceinline__ v2f frag2(const v4f& lo, const v4f& hh, int kk) {
    v2f r;
    if (kk == 0)      { r.x = lo.x; r.y = lo.y; }
    else if (kk == 1) { r.x = lo.z; r.y = lo.w; }
    else if (kk == 2) { r.x = hh.x; r.y = hh.y; }
    else              { r.x = hh.z; r.y = hh.w; }
    return r;
}

// Staging tile is COLUMN-major: element (row,col) at s[col*16 + row].
// C/D tile: VGPR r -> row (r+8*hi), col = lane&15  => contiguous 8 floats/lane.
static __device__ __forceinline__ void store_tile(float* s, const v8f& c, int hi, int ln) {
    v4f lo = {c[0], c[1], c[2], c[3]};
    v4f hh = {c[4], c[5], c[6], c[7]};
    v4f* p = (v4f*)&s[ln * 16 + 8 * hi];
    p[0] = lo;
    p[1] = hh;
}

// A fragment step kk: lane holds row M=ln, K = 4*kk + 2*hi + {0,1}
static __device__ __forceinline__ v2f ldA(const float* s, int kk, int hi, int ln) {
    int k0 = kk * 4 + hi * 2;
    v2f a; a.x = s[k0 * 16 + ln]; a.y = s[(k0 + 1) * 16 + ln];
    return a;
}

// B fragment step kk: lane holds col N=ln, rows K = 4*kk + 2*hi + {0,1}
static __device__ __forceinline__ v2f ldB(const float* s, int kk, int hi, int ln) {
    int k0 = kk * 4 + hi * 2;
    return *(const v2f*)&s[ln * 16 + k0];    // 8B aligned (k0 even)
}

// ---------------- prep: pack constants into fragment layout ----------------
__global__ __launch_bounds__(32)
void prep_kernel(const float* __restrict__ initW,
                 const float* __restrict__ w0, const float* __restrict__ w1,
                 const float* __restrict__ fc1w, const float* __restrict__ fc2w,
                 const float* __restrict__ decw,
                 const float* __restrict__ big, const float* __restrict__ bib,
                 const float* __restrict__ bim, const float* __restrict__ biv,
                 const float* __restrict__ bog, const float* __restrict__ bob,
                 const float* __restrict__ bom, const float* __restrict__ bov,
                 float* __restrict__ ws)
{
    const int lane = threadIdx.x;
    const int hi = lane >> 4, ln = lane & 15;

#pragma unroll
    for (int kk = 0; kk < 4; ++kk) {
#pragma unroll
        for (int j = 0; j < 2; ++j) {
            const int k = kk * 4 + hi * 2 + j;
            const int o = lane * 8 + kk * 2 + j;
            ws[WB_INIT + o] = (k < 13) ? initW[k * 16 + ln] : 0.f;
            ws[WB_FC1 + o]  = fc1w[ln * 16 + k];          // B[k,d]=fc1w[d,k], d=ln
            ws[WB_FC2 + o]  = fc2w[ln * 16 + k];
#pragma unroll
            for (int l = 0; l < 4; ++l) {
                ws[WB_W0 + l * 256 + o]  = w0[l * 256 + ln * 16 + k];
                ws[WB_W1 + l * 256 + o]  = w1[l * 256 + ln * 16 + k];
                ws[WB_DEC + l * 256 + o] = decw[l * 256 + ln * 16 + k];
            }
        }
    }

    if (lane < 16) {
        const int row = lane;
#pragma unroll
        for (int l = 0; l < 4; ++l) {
            float si = 0.f, ti = 0.f, so = 0.f, to = 0.f;
            if (row < 13) {
                si = big[l * 13 + row] * rsqrtf(biv[l * 13 + row] + BN_EPS);
                ti = bib[l * 13 + row] - bim[l * 13 + row] * si;
                so = bog[l * 13 + row] * rsqrtf(bov[l * 13 + row] + BN_EPS);
                to = bob[l * 13 + row] - bom[l * 13 + row] * so;
            }
            ws[BN_SIN  + l * 16 + row] = si;
            ws[BN_TIN  + l * 16 + row] = ti;
            ws[BN_SOUT + l * 16 + row] = so;
            ws[BN_TOUT + l * 16 + row] = to;
        }
    }
}

// ---------------- main kernel: one wave per batch item ----------------
__global__ __launch_bounds__(32)
void gin_vae_kernel(const float* __restrict__ adj,       // (B,4,13,13)
                    const float* __restrict__ epsP,      // (4,)
                    const float* __restrict__ b0,        // (4,16)
                    const float* __restrict__ b1,        // (4,16)
                    const float* __restrict__ fc1b, const float* __restrict__ fc2b,
                    const float* __restrict__ decb,      // (4,16)
                    const float* __restrict__ ws,
                    float* __restrict__ outRecon,        // (B,4,13,13)
                    float* __restrict__ outMu,           // (B,13,16)
                    float* __restrict__ outLv)           // (B,13,16)
{
    __shared__ alignas(16) float s_adjp[4 * 256];   // adj padded to 4 x 16x16
    __shared__ alignas(16) float s_tile[256];       // column-major staging tile

    const int b    = blockIdx.x;
    const int lane = threadIdx.x;     // 0..31
    const int hi   = lane >> 4;
    const int ln   = lane & 15;
    const int grp  = ln >> 2;

    // ---- zero padded adj buffer, then scatter real adj (b128 loads) ----
    {
        v4f z = {0.f, 0.f, 0.f, 0.f};
#pragma unroll
        for (int i = 0; i < 8; ++i) ((v4f*)s_adjp)[lane + 32 * i] = z;
    }
    __syncthreads();
    {
        const v4f* adj4 = (const v4f*)(adj + (size_t)b * 676);   // 2704B, 16B aligned
        for (int i = lane; i < 169; i += 32) {
            v4f v = adj4[i];
            const int base = 4 * i;
#pragma unroll
            for (int e = 0; e < 4; ++e) {
                const int idx = base + e;
                const int ch  = idx / 169;
                const int rem = idx - ch * 169;
                const int n   = rem / 13;
                const int m   = rem - n * 13;
                s_adjp[ch * 256 + n * 16 + m] = v[e];
            }
        }
    }
    __syncthreads();

    // ---- adj A-fragments: clean ds_load_b64, no guards ----
    v2f adjA[4][4];
#pragma unroll
    for (int ch = 0; ch < 4; ++ch)
#pragma unroll
        for (int kk = 0; kk < 4; ++kk)
            adjA[ch][kk] = *(const v2f*)&s_adjp[ch * 256 + ln * 16 + kk * 4 + hi * 2];

    // ---- x = (sum_i adj_i) @ W ----
    v8f xC = zero8();
    {
        const v4f* wp = (const v4f*)(ws + WB_INIT + lane * 8);
        v4f wlo = wp[0], whi = wp[1];
#pragma unroll
        for (int kk = 0; kk < 4; ++kk) {
            v2f a = adjA[0][kk] + adjA[1][kk] + adjA[2][kk] + adjA[3][kk];
            xC = wmma4(a, frag2(wlo, whi, kk), xC);
        }
    }

    __syncthreads();
    store_tile(s_tile, xC, hi, ln);          // s_tile holds current x
    __syncthreads();

    // ---- 4 GIN layers ----
    for (int l = 0; l < 4; ++l) {
        // B-fragments of x
        v2f xb[4];
#pragma unroll
        for (int kk = 0; kk < 4; ++kk) xb[kk] = ldB(s_tile, kk, hi, ln);

        // neighbor[n,c] = sum_m adj_{c>>2}[n,m] * x[m,c]   (mask B per column group)
        v8f nb = zero8();
#pragma unroll
        for (int ch = 0; ch < 4; ++ch) {
            float msk = (grp == ch) ? 1.f : 0.f;
#pragma unroll
            for (int kk = 0; kk < 4; ++kk)
                nb = wmma4(adjA[ch][kk], xb[kk] * msk, nb);
        }

        const float ep = 1.f + epsP[l];
        v8f agg;
#pragma unroll
        for (int r = 0; r < 8; ++r) agg[r] = ep * xC[r] + nb[r];

        __syncthreads();
        store_tile(s_tile, agg, hi, ln);
        __syncthreads();

        // h1 = agg @ w0[l]^T + b0[l]
        v2f aA[4];
#pragma unroll
        for (int kk = 0; kk < 4; ++kk) aA[kk] = ldA(s_tile, kk, hi, ln);

        v8f h1 = zero8();
        {
            const v4f* wp = (const v4f*)(ws + WB_W0 + l * 256 + lane * 8);
            v4f wlo = wp[0], whi = wp[1];
#pragma unroll
            for (int kk = 0; kk < 4; ++kk) h1 = wmma4(aA[kk], frag2(wlo, whi, kk), h1);
        }
        {
            const float bias = b0[l * 16 + ln];
            const v4f* ps = (const v4f*)(ws + BN_SIN + l * 16 + 8 * hi);
            const v4f* pt = (const v4f*)(ws + BN_TIN + l * 16 + 8 * hi);
            v4f s0 = ps[0], s1 = ps[1], t0 = pt[0], t1 = pt[1];
#pragma unroll
            for (int r = 0; r < 8; ++r) {
                float sc = (r < 4) ? s0[r] : s1[r - 4];
                float of = (r < 4) ? t0[r] : t1[r - 4];
                float v  = (h1[r] + bias) * sc + of;       // rows>=13: sc=of=0
                h1[r] = fmaxf(v, 0.01f * v);               // leaky relu
            }
        }

        __syncthreads();
        store_tile(s_tile, h1, hi, ln);
        __syncthreads();

        // out = h1 @ w1[l]^T + b1[l], BN-out, leaky  -> new x
#pragma unroll
        for (int kk = 0; kk < 4; ++kk) aA[kk] = ldA(s_tile, kk, hi, ln);

        v8f oc = zero8();
        {
            const v4f* wp = (const v4f*)(ws + WB_W1 + l * 256 + lane * 8);
            v4f wlo = wp[0], whi = wp[1];
#pragma unroll
            for (int kk = 0; kk < 4; ++kk) oc = wmma4(aA[kk], frag2(wlo, whi, kk), oc);
        }
        {
            const float bias = b1[l * 16 + ln];
            const v4f* ps = (const v4f*)(ws + BN_SOUT + l * 16 + 8 * hi);
            const v4f* pt = (const v4f*)(ws + BN_TOUT + l * 16 + 8 * hi);
            v4f s0 = ps[0], s1 = ps[1], t0 = pt[0], t1 = pt[1];
#pragma unroll
            for (int r = 0; r < 8; ++r) {
                float sc = (r < 4) ? s0[r] : s1[r - 4];
                float of = (r < 4) ? t0[r] : t1[r - 4];
                float v  = (oc[r] + bias) * sc + of;
                xC[r] = fmaxf(v, 0.01f * v);
            }
        }

        __syncthreads();
        store_tile(s_tile, xC, hi, ln);      // next layer's x
        __syncthreads();
    }

    // ---- mu / logvar heads ----
    v2f xA[4];
#pragma unroll
    for (int kk = 0; kk < 4; ++kk) xA[kk] = ldA(s_tile, kk, hi, ln);

    v8f mu = zero8(), lv = zero8();
    {
        const v4f* p1 = (const v4f*)(ws + WB_FC1 + lane * 8);
        const v4f* p2 = (const v4f*)(ws + WB_FC2 + lane * 8);
        v4f a1 = p1[0], b1f = p1[1], a2 = p2[0], b2f = p2[1];
#pragma unroll
        for (int kk = 0; kk < 4; ++kk) {
            mu = wmma4(xA[kk], frag2(a1, b1f, kk), mu);
            lv = wmma4(xA[kk], frag2(a2, b2f, kk), lv);
        }
    }
    {
        const float mb = fc1b[ln], lb = fc2b[ln];
#pragma unroll
        for (int r = 0; r < 8; ++r) { mu[r] += mb; lv[r] += lb; }
    }

    // rows r+8*hi are always valid (<13) for r<5; r>=5 valid only in the hi==0 half.
    {
        float* pm = outMu + (size_t)b * 208 + 8 * hi * 16 + ln;
        float* pl = outLv + (size_t)b * 208 + 8 * hi * 16 + ln;
#pragma unroll
        for (int r = 0; r < 5; ++r) {
            __builtin_nontemporal_store(mu[r], pm + r * 16);
            __builtin_nontemporal_store(lv[r], pl + r * 16);
        }
        if (hi == 0) {
#pragma unroll
            for (int r = 5; r < 8; ++r) {
                __builtin_nontemporal_store(mu[r], pm + r * 16);
                __builtin_nontemporal_store(lv[r], pl + r * 16);
            }
        }
    }

    // ---- decoder: z = mu; temp_k = z @ dec_w[k]^T + dec_b[k]; recon = relu(temp temp^T) ----
    __syncthreads();
    store_tile(s_tile, mu, hi, ln);
    __syncthreads();

    v2f zA[4];
#pragma unroll
    for (int kk = 0; kk < 4; ++kk) zA[kk] = ldA(s_tile, kk, hi, ln);

    for (int kd = 0; kd < 4; ++kd) {
        v8f tc = zero8();
        {
            const v4f* wp = (const v4f*)(ws + WB_DEC + kd * 256 + lane * 8);
            v4f wlo = wp[0], whi = wp[1];
#pragma unroll
            for (int kk = 0; kk < 4; ++kk) tc = wmma4(zA[kk], frag2(wlo, whi, kk), tc);
        }
        const float db = decb[kd * 16 + ln];
#pragma unroll
        for (int r = 0; r < 8; ++r) tc[r] += db;

        __syncthreads();
        store_tile(s_tile, tc, hi, ln);
        __syncthreads();

        // A-frag of temp == B-frag of temp^T elementwise -> one fragment, both operands
        v2f tf[4];
#pragma unroll
        for (int kk = 0; kk < 4; ++kk) tf[kk] = ldA(s_tile, kk, hi, ln);

        v8f rc = zero8();
#pragma unroll
        for (int kk = 0; kk < 4; ++kk) rc = wmma4(tf[kk], tf[kk], rc);

        // columns >=13 never stored; rows valid as in mu store
        if (ln < 13) {
            float* pr = outRecon + (size_t)b * 676 + kd * 169 + (8 * hi) * 13 + ln;
#pragma unroll
            for (int r = 0; r < 5; ++r)
                __builtin_nontemporal_store(fmaxf(rc[r], 0.f), pr + r * 13);
            if (hi == 0) {
#pragma unroll
                for (int r = 5; r < 8; ++r)
                    __builtin_nontemporal_store(fmaxf(rc[r], 0.f), pr + r * 13);
            }
        }
        __syncthreads();   // WAR: s_tile reused next kd iteration
    }
}

extern "C" void kernel_launch(void* const* d_in, const int* in_sizes, int n_in,
                              void* d_out, int out_size, void* d_ws, size_t ws_size,
                              hipStream_t stream) {
    (void)n_in; (void)ws_size; (void)out_size;

    const float* adj   = (const float*)d_in[0];
    const float* initW = (const float*)d_in[1];
    const float* epsP  = (const float*)d_in[2];
    const float* w0    = (const float*)d_in[3];
    const float* b0    = (const float*)d_in[4];
    const float* w1    = (const float*)d_in[5];
    const float* b1    = (const float*)d_in[6];
    const float* big   = (const float*)d_in[7];
    const float* bib   = (const float*)d_in[8];
    const float* bim   = (const float*)d_in[9];
    const float* biv   = (const float*)d_in[10];
    const float* bog   = (const float*)d_in[11];
    const float* bob   = (const float*)d_in[12];
    const float* bom   = (const float*)d_in[13];
    const float* bov   = (const float*)d_in[14];
    const float* fc1w  = (const float*)d_in[15];
    const float* fc1b  = (const float*)d_in[16];
    const float* fc2w  = (const float*)d_in[17];
    const float* fc2b  = (const float*)d_in[18];
    const float* decw  = (const float*)d_in[19];
    const float* decb  = (const float*)d_in[20];

    const int B = in_sizes[0] / 676;   // adj is (B,4,13,13)

    float* ws       = (float*)d_ws;    // needs WS_TOTAL*4 = 16 KB
    float* out      = (float*)d_out;
    float* outRecon = out;                                 // B*676
    float* outMu    = out + (size_t)B * 676;               // B*208
    float* outLv    = outMu + (size_t)B * 208;             // B*208

    prep_kernel<<<1, 32, 0, stream>>>(initW, w0, w1, fc1w, fc2w, decw,
                                      big, bib, bim, biv, bog, bob, bom, bov, ws);

    gin_vae_kernel<<<B, 32, 0, stream>>>(adj, epsP, b0, b1, fc1b, fc2b, decb, ws,
                                         outRecon, outMu, outLv);
}